// AdaptiveEmbeddingI2T_80126909874337
// MI455X (gfx1250) — compile-verified
//
#include <hip/hip_runtime.h>
#include <math.h>

// Problem constants (from reference setup_inputs)
#define D_   1024
#define BI   48
#define BC   48
#define RG   36
#define TT   40
#define EPSB 1e-5f
#define NSPLIT 12          // BC split for deterministic BN partial sums (4 captions each)

typedef __attribute__((ext_vector_type(2))) float v2f;
typedef __attribute__((ext_vector_type(8))) float v8f;

// ---------------------------------------------------------------------------
// K1a: BatchNorm partial sums. grid=(D/256, NSPLIT), block=256.
// Each thread owns one channel d, accumulates sum / sumsq over 4 captions x 40
// tokens (160 coalesced loads, stride D across t). No atomics -> deterministic.
// ---------------------------------------------------------------------------
__global__ void bn_partial_kernel(const float* __restrict__ cap,
                                  float* __restrict__ psum,
                                  float* __restrict__ psq) {
    const int d  = blockIdx.x * 256 + threadIdx.x;
    const int c0 = blockIdx.y * (BC / NSPLIT);
    float s = 0.f, q = 0.f;
    for (int c = c0; c < c0 + (BC / NSPLIT); ++c) {
        const float* p = cap + (size_t)c * TT * D_ + d;
        #pragma unroll 8
        for (int t = 0; t < TT; ++t) {
            float x = p[t * D_];
            s += x;
            q += x * x;
        }
    }
    psum[blockIdx.y * D_ + d] = s;
    psq [blockIdx.y * D_ + d] = q;
}

// K1b: reduce 12 partials per channel -> mean, invstd. grid=D/256, block=256.
__global__ void bn_finalize_kernel(const float* __restrict__ psum,
                                   const float* __restrict__ psq,
                                   float* __restrict__ mean,
                                   float* __restrict__ invstd) {
    const int d = blockIdx.x * 256 + threadIdx.x;
    float s = 0.f, q = 0.f;
    #pragma unroll
    for (int p = 0; p < NSPLIT; ++p) {
        s += psum[p * D_ + d];
        q += psq [p * D_ + d];
    }
    const float inv_n = 1.0f / (float)(BC * TT);
    float mu  = s * inv_n;
    float var = q * inv_n - mu * mu;       // biased variance (torch BN semantics)
    mean[d]   = mu;
    invstd[d] = rsqrtf(var + EPSB);
}

// ---------------------------------------------------------------------------
// K2: img_q[i,d] = mean over 36 regions. grid = BI*D/256, block=256. Coalesced.
// ---------------------------------------------------------------------------
__global__ void img_mean_kernel(const float* __restrict__ img,
                                float* __restrict__ imgq) {
    const int idx = blockIdx.x * 256 + threadIdx.x;      // idx = i*D + d (D==1024)
    const int i = idx >> 10;
    const int d = idx & (D_ - 1);
    const float* p = img + (size_t)i * RG * D_ + d;
    float s = 0.f;
    #pragma unroll 6
    for (int r = 0; r < RG; ++r) s += p[r * D_];
    imgq[idx] = s * (1.0f / (float)RG);
}

// K2b: per-image inverse L2 norm. grid=BI, block=256.
__global__ void img_norm_kernel(const float* __restrict__ imgq,
                                float* __restrict__ inorm) {
    __shared__ float red[256];
    const int i = blockIdx.x, tid = threadIdx.x;
    float s = 0.f;
    for (int d = tid; d < D_; d += 256) {
        float v = imgq[i * D_ + d];
        s += v * v;
    }
    red[tid] = s; __syncthreads();
    for (int st = 128; st > 0; st >>= 1) {
        if (tid < st) red[tid] += red[tid + st];
        __syncthreads();
    }
    if (tid == 0) inorm[i] = rsqrtf(red[0]);
}

// ---------------------------------------------------------------------------
// K3: FiLM GEMMs via V_WMMA_F32_16X16X4_F32 (exact fp32 on the matrix pipe).
//   gamma = imgq @ Wg^T + bg ; beta = imgq @ Wb^T + bb
// One wave32 per 16x16 tile of (M=48, N=1024); both outputs share the A
// fragment, so each K-step issues 2 WMMAs off one A load.
// ISA 7.12.2 fp32 layouts:
//   A 16x4 : lane l in [0,16): M=l, v={K0,K1}; lanes [16,32): M=l&15, v={K2,K3}
//   B 4x16 : lane l in [0,16): N=l, v={K0,K1}; lanes [16,32): N=l&15, v={K2,K3}
//   C 16x16: vgpr r, lane: N=lane&15, M=(lane>>4)*8 + r
// ---------------------------------------------------------------------------
__global__ void film_gemm_kernel(const float* __restrict__ imgq,
                                 const float* __restrict__ Wg,
                                 const float* __restrict__ bg,
                                 const float* __restrict__ Wb,
                                 const float* __restrict__ bb,
                                 float* __restrict__ gp1,     // 1 + gamma
                                 float* __restrict__ beta) {
    const int lane = threadIdx.x;            // 0..31, one wave32 per block
    const int l    = lane & 15;
    const int half = lane >> 4;              // K sub-block {0,1} vs {2,3}
    const int m0 = blockIdx.y * 16;
    const int n0 = blockIdx.x * 16;

    const float* arow  = imgq + (size_t)(m0 + l) * D_;   // A row (M = l)
    const float* bgrow = Wg   + (size_t)(n0 + l) * D_;   // B col n = Wg row n
    const float* bbrow = Wb   + (size_t)(n0 + l) * D_;

    v8f cg = {}; v8f cb = {};
    for (int k0 = 0; k0 < D_; k0 += 4) {
        const int kk = k0 + half * 2;
        v2f a  = *(const v2f*)(arow  + kk);
        v2f bG = *(const v2f*)(bgrow + kk);
        v2f bB = *(const v2f*)(bbrow + kk);
        cg = __builtin_amdgcn_wmma_f32_16x16x4_f32(false, a, false, bG,
                                                   (short)0, cg, false, false);
        cb = __builtin_amdgcn_wmma_f32_16x16x4_f32(false, a, false, bB,
                                                   (short)0, cb, false, false);
    }

    const int   n   = n0 + l;
    const float bgs = bg[n];
    const float bbs = bb[n];
    #pragma unroll
    for (int r = 0; r < 8; ++r) {
        const int m = m0 + half * 8 + r;
        gp1 [(size_t)m * D_ + n] = 1.0f + cg[r] + bgs;
        beta[(size_t)m * D_ + n] = cb[r] + bbs;
    }
}

// ---------------------------------------------------------------------------
// K4: fused FiLM + Fovea + L2 + cosine. grid=(BI,BC), block=256.
// Never materializes the (Bi,Bc,D,T) tensor; per (i,c,d) the 40 token values
// live in registers. cap rereads hit L2 (7.9MB << 192MB). Emits one float.
//   val(d) = max_t( softmax_t(x)*x ) with x = cap_norm*(1+g) + b
//   sims   = (sum_d imgq*val) * inorm_i / sqrt(sum_d val^2)
// ---------------------------------------------------------------------------
__global__ void fovea_sims_kernel(const float* __restrict__ cap,
                                  const float* __restrict__ mean,
                                  const float* __restrict__ invstd,
                                  const float* __restrict__ imgq,
                                  const float* __restrict__ inorm,
                                  const float* __restrict__ gp1,
                                  const float* __restrict__ beta,
                                  float* __restrict__ out) {
    const int i = blockIdx.x, c = blockIdx.y, tid = threadIdx.x;
    __shared__ float red_ss[256];
    __shared__ float red_sd[256];

    float ss = 0.f, sd = 0.f;
    for (int dd = 0; dd < D_; dd += 256) {
        const int d = dd + tid;
        const float mu = mean[d];
        const float is = invstd[d];
        const float g  = gp1 [i * D_ + d];
        const float b  = beta[i * D_ + d];
        const float* capd = cap + (size_t)c * TT * D_ + d;

        float x[TT];
        float mx = -3.0e38f;
        #pragma unroll
        for (int t = 0; t < TT; ++t) {
            float v = (capd[t * D_] - mu) * is * g + b;
            x[t] = v;
            mx = fmaxf(mx, v);
        }
        float s = 0.f, best = -3.0e38f;
        #pragma unroll
        for (int t = 0; t < TT; ++t) {
            float e = __expf(x[t] - mx);
            s += e;
            best = fmaxf(best, e * x[t]);   // 1/s > 0: divide once at the end
        }
        const float val = best / s;
        ss += val * val;
        sd += imgq[i * D_ + d] * val;
    }

    red_ss[tid] = ss;
    red_sd[tid] = sd;
    __syncthreads();
    for (int st = 128; st > 0; st >>= 1) {
        if (tid < st) {
            red_ss[tid] += red_ss[tid + st];
            red_sd[tid] += red_sd[tid + st];
        }
        __syncthreads();
    }
    if (tid == 0)
        out[i * BC + c] = red_sd[0] * inorm[i] * rsqrtf(red_ss[0]);
}

// ---------------------------------------------------------------------------
extern "C" void kernel_launch(void* const* d_in, const int* in_sizes, int n_in,
                              void* d_out, int out_size, void* d_ws, size_t ws_size,
                              hipStream_t stream) {
    const float* img = (const float*)d_in[0];   // (48, 36, 1024)
    const float* cap = (const float*)d_in[1];   // (48, 40, 1024)
    // d_in[2] = lens (int), unused by the reference math
    const float* Wg  = (const float*)d_in[3];   // (1024, 1024)
    const float* bg  = (const float*)d_in[4];   // (1024,)
    const float* Wb  = (const float*)d_in[5];   // (1024, 1024)
    const float* bb  = (const float*)d_in[6];   // (1024,)
    float* out = (float*)d_out;                 // (48, 48)

    // Workspace layout (floats): ~174k floats (~0.7 MB)
    float* ws     = (float*)d_ws;
    float* psum   = ws;                      // NSPLIT * D
    float* psq    = psum   + NSPLIT * D_;    // NSPLIT * D
    float* mean   = psq    + NSPLIT * D_;    // D
    float* invstd = mean   + D_;             // D
    float* imgq   = invstd + D_;             // BI * D
    float* inorm  = imgq   + BI * D_;        // BI (padded to 64)
    float* gp1    = inorm  + 64;             // BI * D
    float* beta   = gp1    + BI * D_;        // BI * D

    bn_partial_kernel  <<<dim3(D_ / 256, NSPLIT), 256, 0, stream>>>(cap, psum, psq);
    bn_finalize_kernel <<<D_ / 256,               256, 0, stream>>>(psum, psq, mean, invstd);
    img_mean_kernel    <<<BI * D_ / 256,          256, 0, stream>>>(img, imgq);
    img_norm_kernel    <<<BI,                     256, 0, stream>>>(imgq, inorm);
    film_gemm_kernel   <<<dim3(D_ / 16, BI / 16),  32, 0, stream>>>(imgq, Wg, bg, Wb, bb, gp1, beta);
    fovea_sims_kernel  <<<dim3(BI, BC),           256, 0, stream>>>(cap, mean, invstd, imgq,
                                                                    inorm, gp1, beta, out);
}